// StructuralGcn_27178553049644
// MI455X (gfx1250) — compile-verified
//
#include <hip/hip_runtime.h>
#include <math.h>

// ---------------------------------------------------------------------------
// StructuralGcn for MI455X (gfx1250, wave32).
// Dense GEMMs use V_WMMA_F32_16X16X4_F32 (fp32 matches the reference exactly);
// the scatter/segment-sum aggregation uses global_atomic_add_f32 (memory
// bound: ~0.7 GB of traffic -> ~40us at 23.3 TB/s; GEMMs are ~1 GFLOP total).
// ---------------------------------------------------------------------------

typedef __attribute__((ext_vector_type(2))) float v2f;
typedef __attribute__((ext_vector_type(8))) float v8f;

#define IN_C  128
#define HID_C 64
#define MID_C 32

// ---------------- degree / normalization ----------------------------------

__global__ void init_deg_kernel(float* __restrict__ deg, int n) {
  int i = blockIdx.x * blockDim.x + threadIdx.x;
  if (i < n) deg[i] = 1.0f;  // self loop contributes 1 to every degree
}

__global__ void edge_deg_kernel(const int* __restrict__ dst,
                                float* __restrict__ deg, int e) {
  int i = blockIdx.x * blockDim.x + threadIdx.x;
  if (i < e) atomicAdd(&deg[dst[i]], 1.0f);
}

__global__ void dinv_kernel(float* __restrict__ degdinv, int n) {
  int i = blockIdx.x * blockDim.x + threadIdx.x;
  if (i < n) {
    float d = degdinv[i];
    degdinv[i] = (d > 0.0f) ? rsqrtf(d) : 0.0f;
  }
}

// ---------------- WMMA fp32 GEMM:  H[M,NOUT] = A[M,K] @ W[K,NOUT] ----------
// One wave32 computes one 16x16 output tile via chained 16x16x4 fp32 WMMAs.
// A-frag:  lane l -> row m0+(l&15), k = kbase, kbase+1 with kbase=2*(l>>4)
// B-frag:  lane l -> col n0+(l&15), same kbase pattern (W staged transposed
//          in LDS with padded stride so each lane reads one float2).
// C/D:     vgpr v, lane l -> (M = v + 8*(l>>4), N = l&15)
// Requires M % 16 == 0 (N_NODES = 50000 = 3125*16).

template <int K, int NOUT>
__global__ __launch_bounds__(256) void gemm_wmma_kernel(
    const float* __restrict__ A, const float* __restrict__ W,
    float* __restrict__ H, int M) {
  constexpr int KPAD = K + 2;          // pad to spread LDS banks
  __shared__ float Bs[16 * KPAD];

  const int lane = threadIdx.x & 31;
  const int wave = threadIdx.x >> 5;   // 0..7
  const int n0   = blockIdx.y * 16;

  // Stage W[:, n0:n0+16] transposed into LDS: Bs[col][k]
  for (int idx = threadIdx.x; idx < 16 * K; idx += 256) {
    const int col = idx / K;
    const int k   = idx - col * K;
    Bs[col * KPAD + k] = W[k * NOUT + n0 + col];
  }
  __syncthreads();

  const int m0 = (blockIdx.x * 8 + wave) * 16;
  if (m0 >= M) return;                 // uniform per-wave exit (EXEC stays all-1)

  const int half = lane >> 4;          // 0 or 1
  const int sub  = lane & 15;
  const float* __restrict__ arow = A + (size_t)(m0 + sub) * K + 2 * half;
  const float* __restrict__ brow = Bs + sub * KPAD + 2 * half;

  v8f acc = {};
#pragma unroll 4
  for (int k = 0; k < K; k += 4) {
    v2f a = *(const v2f*)(arow + k);
    v2f b = *(const v2f*)(brow + k);
    acc = __builtin_amdgcn_wmma_f32_16x16x4_f32(
        /*neg_a=*/false, a, /*neg_b=*/false, b,
        /*c_mod=*/(short)0, acc, /*reuse_a=*/false, /*reuse_b=*/false);
  }

  float* __restrict__ hp = H + (size_t)(m0 + 8 * half) * NOUT + n0 + sub;
#pragma unroll
  for (int v = 0; v < 8; ++v) hp[(size_t)v * NOUT] = acc[v];
}

// ---------------- aggregation (gather-scale-scatter) -----------------------

// out[i,c] = h[i,c] * dinv[i]^2      (self-loop term doubles as initializer)
template <int F>
__global__ void selfloop_init_kernel(const float* __restrict__ h,
                                     const float* __restrict__ dinv,
                                     float* __restrict__ agg, int n) {
  unsigned idx = blockIdx.x * blockDim.x + threadIdx.x;
  if (idx >= (unsigned)n * F) return;
  unsigned i = idx / F;
  float d = dinv[i];
  agg[idx] = h[idx] * d * d;
}

// agg[dst,c] += h[src,c] * dinv[src] * dinv[dst]   (channel in low lane bits
// so each wave issues one coalesced 128B gather / scatter)
template <int F>
__global__ void scatter_kernel(const float* __restrict__ h,
                               const float* __restrict__ dinv,
                               const int* __restrict__ src,
                               const int* __restrict__ dst,
                               float* __restrict__ agg, int e) {
  unsigned idx = blockIdx.x * blockDim.x + threadIdx.x;
  if (idx >= (unsigned)e * F) return;
  unsigned ei = idx / F;
  unsigned c  = idx - ei * F;
  int s = src[ei], d = dst[ei];
  float w = dinv[s] * dinv[d];
  atomicAdd(&agg[(size_t)d * F + c], h[(size_t)s * F + c] * w);
}

template <int F>
__global__ void bias_relu_kernel(float* __restrict__ agg,
                                 const float* __restrict__ b, int n) {
  unsigned idx = blockIdx.x * blockDim.x + threadIdx.x;
  if (idx >= (unsigned)n * F) return;
  float v = agg[idx] + b[idx & (F - 1)];
  agg[idx] = v > 0.0f ? v : 0.0f;
}

// ---------------- layer 3 (NOUT = 1) ---------------------------------------

__global__ void layer3_dot_kernel(const float* __restrict__ h,
                                  const float* __restrict__ W3,
                                  float* __restrict__ h3, int n) {
  int i = blockIdx.x * blockDim.x + threadIdx.x;
  if (i >= n) return;
  float acc = 0.0f;
#pragma unroll
  for (int c = 0; c < MID_C; ++c) acc += h[(size_t)i * MID_C + c] * W3[c];
  h3[i] = acc;
}

__global__ void out_init_kernel(const float* __restrict__ h3,
                                const float* __restrict__ dinv,
                                const float* __restrict__ b3,
                                float* __restrict__ out, int n) {
  int i = blockIdx.x * blockDim.x + threadIdx.x;
  if (i >= n) return;
  float d = dinv[i];
  out[i] = h3[i] * d * d + b3[0];
}

__global__ void out_scatter_kernel(const float* __restrict__ h3,
                                   const float* __restrict__ dinv,
                                   const int* __restrict__ src,
                                   const int* __restrict__ dst,
                                   float* __restrict__ out, int e) {
  int i = blockIdx.x * blockDim.x + threadIdx.x;
  if (i >= e) return;
  int s = src[i], d = dst[i];
  atomicAdd(&out[d], h3[s] * dinv[s] * dinv[d]);
}

// ---------------- launch ----------------------------------------------------

extern "C" void kernel_launch(void* const* d_in, const int* in_sizes, int n_in,
                              void* d_out, int out_size, void* d_ws, size_t ws_size,
                              hipStream_t stream) {
  const float* x  = (const float*)d_in[0];
  const int*   ei = (const int*)d_in[1];
  const float* W1 = (const float*)d_in[2];
  const float* b1 = (const float*)d_in[3];
  const float* W2 = (const float*)d_in[4];
  const float* b2 = (const float*)d_in[5];
  const float* W3 = (const float*)d_in[6];
  const float* b3 = (const float*)d_in[7];
  float* out = (float*)d_out;

  const int n = in_sizes[0] / IN_C;   // 50000 (multiple of 16)
  const int e = in_sizes[1] / 2;      // 800000
  const int* src = ei;
  const int* dst = ei + e;

  // carve scratch (floats): dinv | h1 | agg1 | h2 | agg2 | h3  (~39 MB)
  float* wsf  = (float*)d_ws;
  size_t off  = 0;
  float* dinv = wsf + off; off += (size_t)n;
  float* h1   = wsf + off; off += (size_t)n * HID_C;
  float* agg1 = wsf + off; off += (size_t)n * HID_C;
  float* h2   = wsf + off; off += (size_t)n * MID_C;
  float* agg2 = wsf + off; off += (size_t)n * MID_C;
  float* h3   = wsf + off; off += (size_t)n;
  (void)ws_size; (void)n_in; (void)out_size;

  const int T = 256;
  const unsigned nb  = (n + T - 1) / T;
  const unsigned ebk = (e + T - 1) / T;

  // D^{-1/2} of (A + I)
  init_deg_kernel<<<nb, T, 0, stream>>>(dinv, n);
  edge_deg_kernel<<<ebk, T, 0, stream>>>(dst, dinv, e);
  dinv_kernel<<<nb, T, 0, stream>>>(dinv, n);

  // Layer 1: h1 = x @ W1 ; agg1 = norm-aggregate(h1) ; relu(agg1 + b1)
  {
    dim3 g((n / 16 + 7) / 8, HID_C / 16);
    gemm_wmma_kernel<IN_C, HID_C><<<g, 256, 0, stream>>>(x, W1, h1, n);
  }
  selfloop_init_kernel<HID_C><<<((unsigned)n * HID_C + T - 1) / T, T, 0, stream>>>(h1, dinv, agg1, n);
  scatter_kernel<HID_C><<<((unsigned)e * HID_C + T - 1) / T, T, 0, stream>>>(h1, dinv, src, dst, agg1, e);
  bias_relu_kernel<HID_C><<<((unsigned)n * HID_C + T - 1) / T, T, 0, stream>>>(agg1, b1, n);

  // Layer 2: h2 = agg1 @ W2 ; agg2 = norm-aggregate(h2) ; relu(agg2 + b2)
  {
    dim3 g((n / 16 + 7) / 8, MID_C / 16);
    gemm_wmma_kernel<HID_C, MID_C><<<g, 256, 0, stream>>>(agg1, W2, h2, n);
  }
  selfloop_init_kernel<MID_C><<<((unsigned)n * MID_C + T - 1) / T, T, 0, stream>>>(h2, dinv, agg2, n);
  scatter_kernel<MID_C><<<((unsigned)e * MID_C + T - 1) / T, T, 0, stream>>>(h2, dinv, src, dst, agg2, e);
  bias_relu_kernel<MID_C><<<((unsigned)n * MID_C + T - 1) / T, T, 0, stream>>>(agg2, b2, n);

  // Layer 3: h3 = agg2 @ W3 (32->1) ; out = norm-aggregate(h3) + b3
  layer3_dot_kernel<<<nb, T, 0, stream>>>(agg2, W3, h3, n);
  out_init_kernel<<<nb, T, 0, stream>>>(h3, dinv, b3, out, n);
  out_scatter_kernel<<<ebk, T, 0, stream>>>(h3, dinv, src, dst, out, e);
}